// ComAttention_79491254714603
// MI455X (gfx1250) — compile-verified
//
#include <hip/hip_runtime.h>
#include <hip/hip_bf16.h>
#include <cmath>

// ---------------------------------------------------------------------------
// Problem constants (from the reference): B=8, L=512, H=16, D=1024, DK=64
// ---------------------------------------------------------------------------
constexpr int kB  = 8;
constexpr int kL  = 512;
constexpr int kH  = 16;
constexpr int kD  = 1024;
constexpr int kDK = 64;
constexpr float kScale = 0.125f;           // 1/sqrt(64)

typedef float v2f __attribute__((ext_vector_type(2)));
typedef float v8f __attribute__((ext_vector_type(8)));

// Tile config: 256 threads = 8 waves (wave32). Block tile 128(M) x 64(N),
// K-chunk 32. Waves arranged 4x2; each wave owns a 32x32 macro-tile made of
// a 2x2 grid of 16x16 WMMA tiles (A frags reused across N, B frags across M).
#define BM   128
#define BN   64
#define TKC  32
#define LDP  4     // LDS row padding -> conflict-free column gathers

// ---------------------------------------------------------------------------
// Generic GEMM + bias:  C[M,N] = A[M,K] @ W[K,N] + bias[N]   (fp32 WMMA)
// M=4096, N=K=1024 for every projection in this model.
// ---------------------------------------------------------------------------
__global__ __launch_bounds__(256, 2) void gemm_bias_f32(
    const float* __restrict__ A, const float* __restrict__ W,
    const float* __restrict__ bias, float* __restrict__ C,
    int M, int N, int K)
{
    __shared__ float As[BM][TKC + LDP];   // 128 x 36
    __shared__ float Ws[TKC][BN + LDP];   // 32 x 68

    const int tid   = threadIdx.x;
    const int wave  = tid >> 5;
    const int lane  = tid & 31;
    const int lhalf = lane >> 4;        // 0 | 1
    const int l16   = lane & 15;
    const int bm    = blockIdx.y * BM;
    const int bn    = blockIdx.x * BN;
    const int wm    = (wave >> 1) << 5; // 0,32,64,96
    const int wn    = (wave & 1) << 5;  // 0,32

    v8f acc00 = {0.f,0.f,0.f,0.f,0.f,0.f,0.f,0.f};
    v8f acc01 = acc00, acc10 = acc00, acc11 = acc00;

    for (int k0 = 0; k0 < K; k0 += TKC) {
        // Stage A tile 128x32 (four float4 per thread)
        #pragma unroll
        for (int it = 0; it < 4; ++it) {
            const int idx = tid + it * 256;
            const int r = idx >> 3;
            const int c = (idx & 7) << 2;
            const float4 v = *reinterpret_cast<const float4*>(
                &A[(size_t)(bm + r) * K + k0 + c]);
            *reinterpret_cast<float4*>(&As[r][c]) = v;
        }
        // Stage W tile 32x64 (two float4 per thread)
        #pragma unroll
        for (int it = 0; it < 2; ++it) {
            const int idx = tid + it * 256;
            const int r = idx >> 4;
            const int c = (idx & 15) << 2;
            const float4 v = *reinterpret_cast<const float4*>(
                &W[(size_t)(k0 + r) * N + bn + c]);
            *reinterpret_cast<float4*>(&Ws[r][c]) = v;
        }
        __syncthreads();

        if (k0 + TKC < K)   // CDNA5 global_prefetch_b8 of next A chunk
            __builtin_prefetch(&A[(size_t)(bm + (tid >> 3)) * K + k0 + TKC], 0, 3);

        #pragma unroll
        for (int kk = 0; kk < TKC; kk += 4) {
            v2f a0, a1, b0, b1;
            a0.x = As[wm +      l16][kk + 2 * lhalf + 0];
            a0.y = As[wm +      l16][kk + 2 * lhalf + 1];
            a1.x = As[wm + 16 + l16][kk + 2 * lhalf + 0];
            a1.y = As[wm + 16 + l16][kk + 2 * lhalf + 1];
            b0.x = Ws[kk + 2 * lhalf + 0][wn +      l16];
            b0.y = Ws[kk + 2 * lhalf + 1][wn +      l16];
            b1.x = Ws[kk + 2 * lhalf + 0][wn + 16 + l16];
            b1.y = Ws[kk + 2 * lhalf + 1][wn + 16 + l16];
            acc00 = __builtin_amdgcn_wmma_f32_16x16x4_f32(false, a0, false, b0, (short)0, acc00, false, false);
            acc01 = __builtin_amdgcn_wmma_f32_16x16x4_f32(false, a0, false, b1, (short)0, acc01, false, false);
            acc10 = __builtin_amdgcn_wmma_f32_16x16x4_f32(false, a1, false, b0, (short)0, acc10, false, false);
            acc11 = __builtin_amdgcn_wmma_f32_16x16x4_f32(false, a1, false, b1, (short)0, acc11, false, false);
        }
        __syncthreads();
    }

    const int col0 = bn + wn + l16;
    const int col1 = col0 + 16;
    const float bv0 = bias[col0];
    const float bv1 = bias[col1];
    #pragma unroll
    for (int r = 0; r < 8; ++r) {
        const int row0 = bm + wm + (lhalf << 3) + r;
        const int row1 = row0 + 16;
        C[(size_t)row0 * N + col0] = acc00[r] + bv0;
        C[(size_t)row0 * N + col1] = acc01[r] + bv1;
        C[(size_t)row1 * N + col0] = acc10[r] + bv0;
        C[(size_t)row1 * N + col1] = acc11[r] + bv1;
    }
}

// ---------------------------------------------------------------------------
// Attention scores: S[b,h,i,j] = scale * <q[b,i,h,:], k[b,j,h,:]>
// optionally masked_fill(mask[b,i,j]==0, -1e9).   Q,K stored [B,L,D].
// One (b,h) per blockIdx.z; M=N=512, Kdim=64.
// ---------------------------------------------------------------------------
__global__ __launch_bounds__(256, 2) void attn_scores_f32(
    const float* __restrict__ Q, const float* __restrict__ Kmat,
    const float* __restrict__ mask, float* __restrict__ S, float scale)
{
    __shared__ float Qs[BM][TKC + LDP];     // rows i, cols d (128 x 36)
    __shared__ float Ks[BN][TKC + LDP];     // rows j, cols d (64 x 36)

    const int tid   = threadIdx.x;
    const int wave  = tid >> 5;
    const int lane  = tid & 31;
    const int lhalf = lane >> 4;
    const int l16   = lane & 15;
    const int bh    = blockIdx.z;
    const int b     = bh / kH;
    const int h     = bh - b * kH;
    const int bm    = blockIdx.y * BM;      // i tile
    const int bn    = blockIdx.x * BN;      // j tile
    const int wm    = (wave >> 1) << 5;
    const int wn    = (wave & 1) << 5;
    const size_t base = (size_t)b * kL * kD + (size_t)h * kDK;

    v8f acc00 = {0.f,0.f,0.f,0.f,0.f,0.f,0.f,0.f};
    v8f acc01 = acc00, acc10 = acc00, acc11 = acc00;

    for (int k0 = 0; k0 < kDK; k0 += TKC) {   // 2 iterations
        #pragma unroll
        for (int it = 0; it < 4; ++it) {
            const int idx = tid + it * 256;
            const int r = idx >> 3;
            const int c = (idx & 7) << 2;
            const float4 v = *reinterpret_cast<const float4*>(
                &Q[base + (size_t)(bm + r) * kD + k0 + c]);
            *reinterpret_cast<float4*>(&Qs[r][c]) = v;
        }
        #pragma unroll
        for (int it = 0; it < 2; ++it) {
            const int idx = tid + it * 256;
            const int r = idx >> 3;
            const int c = (idx & 7) << 2;
            const float4 v = *reinterpret_cast<const float4*>(
                &Kmat[base + (size_t)(bn + r) * kD + k0 + c]);
            *reinterpret_cast<float4*>(&Ks[r][c]) = v;
        }
        __syncthreads();

        #pragma unroll
        for (int kk = 0; kk < TKC; kk += 4) {
            v2f a0, a1, b0, b1;
            a0.x = Qs[wm +      l16][kk + 2 * lhalf + 0];
            a0.y = Qs[wm +      l16][kk + 2 * lhalf + 1];
            a1.x = Qs[wm + 16 + l16][kk + 2 * lhalf + 0];
            a1.y = Qs[wm + 16 + l16][kk + 2 * lhalf + 1];
            // B[kdim][n] = K^T  ->  Ks[n][kdim]
            b0.x = Ks[wn +      l16][kk + 2 * lhalf + 0];
            b0.y = Ks[wn +      l16][kk + 2 * lhalf + 1];
            b1.x = Ks[wn + 16 + l16][kk + 2 * lhalf + 0];
            b1.y = Ks[wn + 16 + l16][kk + 2 * lhalf + 1];
            acc00 = __builtin_amdgcn_wmma_f32_16x16x4_f32(false, a0, false, b0, (short)0, acc00, false, false);
            acc01 = __builtin_amdgcn_wmma_f32_16x16x4_f32(false, a0, false, b1, (short)0, acc01, false, false);
            acc10 = __builtin_amdgcn_wmma_f32_16x16x4_f32(false, a1, false, b0, (short)0, acc10, false, false);
            acc11 = __builtin_amdgcn_wmma_f32_16x16x4_f32(false, a1, false, b1, (short)0, acc11, false, false);
        }
        __syncthreads();
    }

    const size_t srow = (size_t)bh * kL;
    #pragma unroll
    for (int mi = 0; mi < 2; ++mi) {
        #pragma unroll
        for (int ni = 0; ni < 2; ++ni) {
            const v8f acc = (mi == 0) ? (ni == 0 ? acc00 : acc01)
                                      : (ni == 0 ? acc10 : acc11);
            const int col = bn + wn + 16 * ni + l16;
            #pragma unroll
            for (int r = 0; r < 8; ++r) {
                const int row = bm + wm + 16 * mi + (lhalf << 3) + r;
                float s = acc[r] * scale;
                if (mask != nullptr) {
                    const float mv = mask[((size_t)b * kL + row) * kL + col];
                    if (mv == 0.0f) s = -1e9f;
                }
                S[(srow + row) * kL + col] = s;
            }
        }
    }
}

// ---------------------------------------------------------------------------
// Attention apply: ctx[b,i,h,:] = sum_j P[b,h,i,j] * v[b,j,h,:]
// P is [B,H,L,L]; V, ctx are [B,L,D]. Per (b,h): M=512, N=64 (=BN), Kdim=512.
// ---------------------------------------------------------------------------
__global__ __launch_bounds__(256, 2) void attn_apply_f32(
    const float* __restrict__ P, const float* __restrict__ V,
    float* __restrict__ ctx)
{
    __shared__ float Ps[BM][TKC + LDP];
    __shared__ float Vs[TKC][BN + LDP];

    const int tid   = threadIdx.x;
    const int wave  = tid >> 5;
    const int lane  = tid & 31;
    const int lhalf = lane >> 4;
    const int l16   = lane & 15;
    const int bh    = blockIdx.z;
    const int b     = bh / kH;
    const int h     = bh - b * kH;
    const int bm    = blockIdx.y * BM;      // i tile (512/128)
    const int wm    = (wave >> 1) << 5;
    const int wn    = (wave & 1) << 5;
    const size_t pbase = (size_t)bh * kL * kL;
    const size_t vbase = (size_t)b * kL * kD + (size_t)h * kDK;

    v8f acc00 = {0.f,0.f,0.f,0.f,0.f,0.f,0.f,0.f};
    v8f acc01 = acc00, acc10 = acc00, acc11 = acc00;

    for (int k0 = 0; k0 < kL; k0 += TKC) {   // 16 iterations over j
        #pragma unroll
        for (int it = 0; it < 4; ++it) {
            const int idx = tid + it * 256;
            const int r = idx >> 3;
            const int c = (idx & 7) << 2;
            const float4 v = *reinterpret_cast<const float4*>(
                &P[pbase + (size_t)(bm + r) * kL + k0 + c]);
            *reinterpret_cast<float4*>(&Ps[r][c]) = v;
        }
        #pragma unroll
        for (int it = 0; it < 2; ++it) {
            const int idx = tid + it * 256;
            const int r = idx >> 4;
            const int c = (idx & 15) << 2;
            const float4 v = *reinterpret_cast<const float4*>(
                &V[vbase + (size_t)(k0 + r) * kD + c]);
            *reinterpret_cast<float4*>(&Vs[r][c]) = v;
        }
        __syncthreads();

        #pragma unroll
        for (int kk = 0; kk < TKC; kk += 4) {
            v2f a0, a1, b0, b1;
            a0.x = Ps[wm +      l16][kk + 2 * lhalf + 0];
            a0.y = Ps[wm +      l16][kk + 2 * lhalf + 1];
            a1.x = Ps[wm + 16 + l16][kk + 2 * lhalf + 0];
            a1.y = Ps[wm + 16 + l16][kk + 2 * lhalf + 1];
            b0.x = Vs[kk + 2 * lhalf + 0][wn +      l16];
            b0.y = Vs[kk + 2 * lhalf + 1][wn +      l16];
            b1.x = Vs[kk + 2 * lhalf + 0][wn + 16 + l16];
            b1.y = Vs[kk + 2 * lhalf + 1][wn + 16 + l16];
            acc00 = __builtin_amdgcn_wmma_f32_16x16x4_f32(false, a0, false, b0, (short)0, acc00, false, false);
            acc01 = __builtin_amdgcn_wmma_f32_16x16x4_f32(false, a0, false, b1, (short)0, acc01, false, false);
            acc10 = __builtin_amdgcn_wmma_f32_16x16x4_f32(false, a1, false, b0, (short)0, acc10, false, false);
            acc11 = __builtin_amdgcn_wmma_f32_16x16x4_f32(false, a1, false, b1, (short)0, acc11, false, false);
        }
        __syncthreads();
    }

    #pragma unroll
    for (int mi = 0; mi < 2; ++mi) {
        #pragma unroll
        for (int ni = 0; ni < 2; ++ni) {
            const v8f acc = (mi == 0) ? (ni == 0 ? acc00 : acc01)
                                      : (ni == 0 ? acc10 : acc11);
            const int col = wn + 16 * ni + l16;
            #pragma unroll
            for (int r = 0; r < 8; ++r) {
                const int row = bm + wm + 16 * mi + (lhalf << 3) + r;
                ctx[((size_t)b * kL + row) * kD + h * kDK + col] = acc[r];
            }
        }
    }
}

// ---------------------------------------------------------------------------
// Row softmax over rows of length 512, in-place. One block (256 thr) per row.
// ---------------------------------------------------------------------------
__global__ __launch_bounds__(256) void softmax_rows_f32(float* __restrict__ S)
{
    __shared__ float red[256];
    float* p = S + (size_t)blockIdx.x * kL;
    const int t = threadIdx.x;
    const float v0 = p[t];
    const float v1 = p[t + 256];

    red[t] = fmaxf(v0, v1);
    __syncthreads();
    for (int s = 128; s > 0; s >>= 1) {
        if (t < s) red[t] = fmaxf(red[t], red[t + s]);
        __syncthreads();
    }
    const float m = red[0];
    __syncthreads();

    const float e0 = __expf(v0 - m);
    const float e1 = __expf(v1 - m);
    red[t] = e0 + e1;
    __syncthreads();
    for (int s = 128; s > 0; s >>= 1) {
        if (t < s) red[t] += red[t + s];
        __syncthreads();
    }
    const float inv = 1.0f / red[0];
    p[t]       = e0 * inv;
    p[t + 256] = e1 * inv;
}

// ---------------------------------------------------------------------------
// Fold fw3 @ fw2 @ fw1 (no nonlinearity between convs) into weff[16], beff.
// Single block of 32 threads.
// ---------------------------------------------------------------------------
__global__ void fold_fusion_f32(
    const float* __restrict__ fw1, const float* __restrict__ fb1,
    const float* __restrict__ fw2, const float* __restrict__ fb2,
    const float* __restrict__ fw3, const float* __restrict__ fb3,
    float* __restrict__ wfold /* [0..15]=weff, [16]=beff */)
{
    const int c = threadIdx.x;
    if (c < 16) {
        float acc = 0.f;
        for (int o = 0; o < 16; ++o) {
            float w2w1 = 0.f;
            for (int m = 0; m < 32; ++m)
                w2w1 += fw2[o * 32 + m] * fw1[m * 16 + c];
            acc += fw3[o] * w2w1;
        }
        wfold[c] = acc;
    } else if (c == 16) {
        float acc = fb3[0];
        for (int o = 0; o < 16; ++o) {
            float t = fb2[o];
            for (int m = 0; m < 32; ++m)
                t += fw2[o * 32 + m] * fb1[m];
            acc += fw3[o] * t;
        }
        wfold[16] = acc;
    }
}

// ---------------------------------------------------------------------------
// Build p/n masks: h = dot(weff, scores[:,c,i,j]) + beff; s = sigmoid(h);
// pmask = (s>0.5 ? s : 0)*dm ; nmask = (s<=0.5 ? s : 0)*dm.
// ---------------------------------------------------------------------------
__global__ __launch_bounds__(256) void build_masks_f32(
    const float* __restrict__ S0, const float* __restrict__ wfold,
    const float* __restrict__ dmask,
    float* __restrict__ pmask, float* __restrict__ nmask)
{
    const size_t idx = (size_t)blockIdx.x * 256 + threadIdx.x;  // over B*L*L
    const size_t ll = (size_t)kL * kL;
    const int b = (int)(idx / ll);
    const size_t r = idx - (size_t)b * ll;

    float h = wfold[16];
    #pragma unroll
    for (int c = 0; c < kH; ++c)
        h += wfold[c] * S0[((size_t)(b * kH + c)) * ll + r];

    const float s  = 1.0f / (1.0f + __expf(-h));
    const float dm = dmask[idx];
    pmask[idx] = (s >  0.5f ? s : 0.0f) * dm;
    nmask[idx] = (s <= 0.5f ? s : 0.0f) * dm;
}

// ---------------------------------------------------------------------------
// Final gated blend: softmax over 2-stack == sigmoid(ep-en).
// ---------------------------------------------------------------------------
__global__ __launch_bounds__(256) void combine_gate_f32(
    const float* __restrict__ vp, const float* __restrict__ vn,
    const float* __restrict__ ep, const float* __restrict__ en,
    float* __restrict__ out)
{
    const size_t i = (size_t)blockIdx.x * 256 + threadIdx.x;
    const float g = 1.0f / (1.0f + __expf(-(ep[i] - en[i])));
    out[i] = vp[i] * g + vn[i] * (1.0f - g);
}

// ---------------------------------------------------------------------------
// Host-side orchestration
// ---------------------------------------------------------------------------
extern "C" void kernel_launch(void* const* d_in, const int* in_sizes, int n_in,
                              void* d_out, int out_size, void* d_ws, size_t ws_size,
                              hipStream_t stream)
{
    (void)in_sizes; (void)n_in; (void)out_size; (void)ws_size;

    const float* feature   = (const float*)d_in[0];
    const float* data_mask = (const float*)d_in[1];
    const float* Wq  = (const float*)d_in[2];
    const float* bq  = (const float*)d_in[3];
    const float* Wk  = (const float*)d_in[4];
    const float* bk  = (const float*)d_in[5];
    const float* fw1 = (const float*)d_in[6];
    const float* fb1 = (const float*)d_in[7];
    const float* fw2 = (const float*)d_in[8];
    const float* fb2 = (const float*)d_in[9];
    const float* fw3 = (const float*)d_in[10];
    const float* fb3 = (const float*)d_in[11];
    const float* p_Wq = (const float*)d_in[12];
    const float* p_bq = (const float*)d_in[13];
    const float* p_Wk = (const float*)d_in[14];
    const float* p_bk = (const float*)d_in[15];
    const float* p_Wv = (const float*)d_in[16];
    const float* p_bv = (const float*)d_in[17];
    const float* p_Wo = (const float*)d_in[18];
    const float* p_bo = (const float*)d_in[19];
    const float* n_Wq = (const float*)d_in[20];
    const float* n_bq = (const float*)d_in[21];
    const float* n_Wk = (const float*)d_in[22];
    const float* n_bk = (const float*)d_in[23];
    const float* n_Wv = (const float*)d_in[24];
    const float* n_bv = (const float*)d_in[25];
    const float* n_Wo = (const float*)d_in[26];
    const float* n_bo = (const float*)d_in[27];
    const float* Wvp = (const float*)d_in[28];
    const float* bvp = (const float*)d_in[29];
    const float* Wvn = (const float*)d_in[30];
    const float* bvn = (const float*)d_in[31];
    const float* Wgp = (const float*)d_in[32];
    const float* bgp = (const float*)d_in[33];
    // NOTE: source uses gate_p (Wgp/bgp) for BOTH branches; Wgn/bgn unused.

    constexpr size_t nBLD  = (size_t)kB * kL * kD;            // 4,194,304
    constexpr size_t nBLL  = (size_t)kB * kL * kL;            // 2,097,152
    constexpr size_t nBHLL = (size_t)kB * kH * kL * kL;       // 33,554,432
    constexpr int M = kB * kL;                                // 4096

    float* ws   = (float*)d_ws;
    float* bufA = ws;                  // q / later vp
    float* bufB = bufA + nBLD;         // k / later vn
    float* bufC = bufB + nBLD;         // v / later ep
    float* bufD = bufC + nBLD;         // ctx / later en
    float* bufE = bufD + nBLD;         // p branch output
    float* bufF = bufE + nBLD;         // n branch output
    float* pmk  = bufF + nBLD;         // p mask [B,L,L]
    float* nmk  = pmk + nBLL;          // n mask [B,L,L]
    float* wfold = nmk + nBLL;         // folded fusion weights (17 floats)

    float* outF  = (float*)d_out;      // [B,L,D]
    float* pmOut = outF + nBLD;        // [B,H,L,L]
    float* nmOut = pmOut + nBHLL;      // [B,H,L,L]

    const dim3 blk(256);
    const dim3 gGemm(kD / BN, M / BM);            // (16, 32)
    const dim3 gScore(kL / BN, kL / BM, kB * kH); // (8, 4, 128)
    const dim3 gAttnV(kDK / BN, kL / BM, kB * kH);// (1, 4, 128)
    const dim3 gSmax(kB * kH * kL);               // 65536 rows
    const dim3 gMask((unsigned)(nBLL / 256));
    const dim3 gComb((unsigned)(nBLD / 256));

    // 1) top-level q, k projections
    gemm_bias_f32<<<gGemm, blk, 0, stream>>>(feature, Wq, bq, bufA, M, kD, kD);
    gemm_bias_f32<<<gGemm, blk, 0, stream>>>(feature, Wk, bk, bufB, M, kD, kD);

    // 2) initial (unmasked, scaled) scores -> pm output region (scratch reuse)
    attn_scores_f32<<<gScore, blk, 0, stream>>>(bufA, bufB, nullptr, pmOut, kScale);

    // 3) fold the linear 16->32->16->1 conv stack, then build p/n masks
    fold_fusion_f32<<<dim3(1), dim3(32), 0, stream>>>(fw1, fb1, fw2, fb2, fw3, fb3, wfold);
    build_masks_f32<<<gMask, blk, 0, stream>>>(pmOut, wfold, data_mask, pmk, nmk);

    // 4) p-branch MHA
    gemm_bias_f32<<<gGemm, blk, 0, stream>>>(feature, p_Wq, p_bq, bufA, M, kD, kD);
    gemm_bias_f32<<<gGemm, blk, 0, stream>>>(feature, p_Wk, p_bk, bufB, M, kD, kD);
    gemm_bias_f32<<<gGemm, blk, 0, stream>>>(feature, p_Wv, p_bv, bufC, M, kD, kD);
    attn_scores_f32<<<gScore, blk, 0, stream>>>(bufA, bufB, pmk, pmOut, kScale);
    softmax_rows_f32<<<gSmax, blk, 0, stream>>>(pmOut);           // final pm
    attn_apply_f32<<<gAttnV, blk, 0, stream>>>(pmOut, bufC, bufD);
    gemm_bias_f32<<<gGemm, blk, 0, stream>>>(bufD, p_Wo, p_bo, bufE, M, kD, kD);

    // 5) n-branch MHA
    gemm_bias_f32<<<gGemm, blk, 0, stream>>>(feature, n_Wq, n_bq, bufA, M, kD, kD);
    gemm_bias_f32<<<gGemm, blk, 0, stream>>>(feature, n_Wk, n_bk, bufB, M, kD, kD);
    gemm_bias_f32<<<gGemm, blk, 0, stream>>>(feature, n_Wv, n_bv, bufC, M, kD, kD);
    attn_scores_f32<<<gScore, blk, 0, stream>>>(bufA, bufB, nmk, nmOut, kScale);
    softmax_rows_f32<<<gSmax, blk, 0, stream>>>(nmOut);           // final nm
    attn_apply_f32<<<gAttnV, blk, 0, stream>>>(nmOut, bufC, bufD);
    gemm_bias_f32<<<gGemm, blk, 0, stream>>>(bufD, n_Wo, n_bo, bufF, M, kD, kD);

    // 6) value / gate projections (gate uses Wgp for both per source)
    gemm_bias_f32<<<gGemm, blk, 0, stream>>>(bufE, Wvp, bvp, bufA, M, kD, kD);  // vp
    gemm_bias_f32<<<gGemm, blk, 0, stream>>>(bufF, Wvn, bvn, bufB, M, kD, kD);  // vn
    gemm_bias_f32<<<gGemm, blk, 0, stream>>>(bufE, Wgp, bgp, bufC, M, kD, kD);  // ep
    gemm_bias_f32<<<gGemm, blk, 0, stream>>>(bufF, Wgp, bgp, bufD, M, kD, kD);  // en

    // 7) gated blend -> out_feature
    combine_gate_f32<<<gComb, blk, 0, stream>>>(bufA, bufB, bufC, bufD, outF);

    (void)Wvn; (void)bvn; // silence unused warnings if any path changes
}